// MULTIGIN_16810501996621
// MI455X (gfx1250) — compile-verified
//
#include <hip/hip_runtime.h>

// MULTIGIN on gfx1250: fp32 GIN via V_WMMA_F32_16X16X4_F32, edge scatter via
// GLOBAL_ATOMIC_ADD_F32 (node feature table is L2-resident: 25.6MB << 192MB).
// Weights staged in LDS in k-pair-interleaved layout so every WMMA B fragment
// is a single aligned ds_load_b64 into an even VGPR pair (no repack movs).

typedef __attribute__((ext_vector_type(2))) float v2f;
typedef __attribute__((ext_vector_type(8))) float v8f;

#define D         64
#define DCAT      256
#define NCLS      40
#define MLP_WAVES 4
#define WPR       160  // dwords per k-pair row (128 data + 32 pad -> conflict-free)
#define CPR       96   // classifier: 48 cols * 2 = 96; 96 = 64+32 self-pads banks
#define ZSTR      68   // Z-tile row stride: b64 A-reads hit all 64 banks exactly

__device__ __forceinline__ v8f wmma4(v2f a, v2f b, v8f c) {
  // 8 args: (neg_a, A, neg_b, B, c_mod, C, reuse_a, reuse_b)
  return __builtin_amdgcn_wmma_f32_16x16x4_f32(false, a, false, b, (short)0, c,
                                               false, false);
}

// ---- copy x into columns [0,64) of h_cat ([N,256], lives in d_out) ----
__global__ void copy_x_kernel(const float* __restrict__ x,
                              float* __restrict__ hcat, int total) {
  int gid = blockIdx.x * blockDim.x + threadIdx.x;
  if (gid >= total) return;
  int i = gid >> 6, c = gid & 63;
  hcat[(size_t)i * DCAT + c] = x[gid];
}

// ---- agg = h_l  (folds the "+ h" of GINConv eps=0) ----
__global__ void agg_init_kernel(float* __restrict__ agg,
                                const float* __restrict__ hcat, int col_off,
                                int total) {
  int gid = blockIdx.x * blockDim.x + threadIdx.x;
  if (gid >= total) return;
  int i = gid >> 6, c = gid & 63;
  agg[gid] = hcat[(size_t)i * DCAT + col_off + c];
}

// ---- agg[dst] += h_l[src] : 16 threads/edge, float4 gather + 4 f32 atomics ----
__global__ void scatter_kernel(float* __restrict__ agg,
                               const float* __restrict__ hcat, int col_off,
                               const int* __restrict__ src,
                               const int* __restrict__ dst, int nedge) {
  int gid = blockIdx.x * blockDim.x + threadIdx.x;
  int e = gid >> 4;
  if (e >= nedge) return;
  int q = (gid & 15) << 2;
  int s = src[e];
  int d = dst[e];
  const float4 v = *(const float4*)(hcat + (size_t)s * DCAT + col_off + q);
  float* p = agg + (size_t)d * D + q;
  atomicAdd(p + 0, v.x);
  atomicAdd(p + 1, v.y);
  atomicAdd(p + 2, v.z);
  atomicAdd(p + 3, v.w);
}

// ---- h_{l+1} = relu(agg@W1 + b1)@W2 + b2, one 16-row tile per wave ----
__global__ __launch_bounds__(MLP_WAVES * 32) void gin_mlp_kernel(
    const float* __restrict__ agg, const float* __restrict__ W1,
    const float* __restrict__ b1, const float* __restrict__ W2,
    const float* __restrict__ b2, float* __restrict__ hcat, int col_off,
    int nnodes) {
  __shared__ float sW1[32 * WPR];              // 20 KB, k-pair interleaved
  __shared__ float sW2[32 * WPR];              // 20 KB
  __shared__ float sZ[MLP_WAVES][16 * ZSTR];   // 17 KB
  for (int i = threadIdx.x; i < D * D; i += blockDim.x) {
    const int k = i >> 6, c = i & 63;
    const int idx = (k >> 1) * WPR + c * 2 + (k & 1);
    sW1[idx] = W1[i];
    sW2[idx] = W2[i];
  }
  __syncthreads();

  const int wave = threadIdx.x >> 5;
  const int lane = threadIdx.x & 31;
  const int row0 = (blockIdx.x * MLP_WAVES + wave) * 16;
  if (row0 + 16 > nnodes) return;  // wave-uniform; EXEC stays all-ones for WMMA

  const int m  = lane & 15;   // A-row / D-column within tile
  const int hi = lane >> 4;   // half-wave selects K pair (A/B) and M+8 (C/D)
  const float* aRow = agg + (size_t)(row0 + m) * D;

  const v8f zero = {0.f, 0.f, 0.f, 0.f, 0.f, 0.f, 0.f, 0.f};
  v8f acc[4] = {zero, zero, zero, zero};

  // GEMM1: [16x64] @ [64x64], K in steps of 4, 4 N-tiles of 16
  for (int k0 = 0; k0 < D; k0 += 4) {
    const int ka = k0 + hi * 2;
    const v2f a = *(const v2f*)(aRow + ka);                 // global_load_b64
    const float* wrow = &sW1[(ka >> 1) * WPR];
#pragma unroll
    for (int nt = 0; nt < 4; ++nt) {
      const v2f b = *(const v2f*)(wrow + (nt * 16 + m) * 2); // ds_load_b64
      acc[nt] = wmma4(a, b, acc[nt]);
    }
  }

  // bias + ReLU, reshape D-layout -> A-layout via per-wave LDS tile
  float* z = &sZ[wave][0];
#pragma unroll
  for (int nt = 0; nt < 4; ++nt) {
    const int col = nt * 16 + m;
    const float bias = b1[col];
#pragma unroll
    for (int v = 0; v < 8; ++v) {
      z[(v + hi * 8) * ZSTR + col] = fmaxf(acc[nt][v] + bias, 0.f);
    }
  }
  __builtin_amdgcn_wave_barrier();  // same-wave LDS exchange; DS ops in-order

  // GEMM2: [16x64] @ [64x64]
  v8f acc2[4] = {zero, zero, zero, zero};
  for (int k0 = 0; k0 < D; k0 += 4) {
    const int ka = k0 + hi * 2;
    const v2f a = *(const v2f*)(z + m * ZSTR + ka);          // ds_load_b64
    const float* wrow = &sW2[(ka >> 1) * WPR];
#pragma unroll
    for (int nt = 0; nt < 4; ++nt) {
      const v2f b = *(const v2f*)(wrow + (nt * 16 + m) * 2); // ds_load_b64
      acc2[nt] = wmma4(a, b, acc2[nt]);
    }
  }

  // store h_{l+1} into its column slice of h_cat
#pragma unroll
  for (int nt = 0; nt < 4; ++nt) {
    const int col = nt * 16 + m;
    const float bias = b2[col];
#pragma unroll
    for (int v = 0; v < 8; ++v) {
      hcat[(size_t)(row0 + v + hi * 8) * DCAT + col_off + col] =
          acc2[nt][v] + bias;
    }
  }
}

// ---- logits = h_cat[N,256] @ Wlin[256,40] + blin ----
__global__ __launch_bounds__(128) void classifier_kernel(
    const float* __restrict__ hcat, const float* __restrict__ Wlin,
    const float* __restrict__ blin, float* __restrict__ logits, int nnodes) {
  __shared__ float sW[128 * CPR];  // 48 KB: 48 zero-padded cols, pair-interleaved
  for (int i = threadIdx.x; i < DCAT * 48; i += blockDim.x) {
    const int k = i / 48, c = i - k * 48;
    sW[(k >> 1) * CPR + c * 2 + (k & 1)] = (c < NCLS) ? Wlin[k * NCLS + c] : 0.f;
  }
  __syncthreads();

  const int wave = threadIdx.x >> 5;
  const int lane = threadIdx.x & 31;
  const int row0 = (blockIdx.x * 4 + wave) * 16;
  if (row0 + 16 > nnodes) return;

  const int m  = lane & 15;
  const int hi = lane >> 4;
  const float* aRow = hcat + (size_t)(row0 + m) * DCAT;

  const v8f zero = {0.f, 0.f, 0.f, 0.f, 0.f, 0.f, 0.f, 0.f};
  v8f acc[3] = {zero, zero, zero};

  for (int k0 = 0; k0 < DCAT; k0 += 4) {
    const int ka = k0 + hi * 2;
    const v2f a = *(const v2f*)(aRow + ka);                  // global_load_b64
    const float* wrow = &sW[(ka >> 1) * CPR];
#pragma unroll
    for (int nt = 0; nt < 3; ++nt) {
      const v2f b = *(const v2f*)(wrow + (nt * 16 + m) * 2); // ds_load_b64
      acc[nt] = wmma4(a, b, acc[nt]);
    }
  }

#pragma unroll
  for (int nt = 0; nt < 3; ++nt) {
    const int col = nt * 16 + m;
    if (col < NCLS) {  // divergence only after all WMMAs are done
      const float bias = blin[col];
#pragma unroll
      for (int v = 0; v < 8; ++v) {
        logits[(size_t)(row0 + v + hi * 8) * NCLS + col] = acc[nt][v] + bias;
      }
    }
  }
}

extern "C" void kernel_launch(void* const* d_in, const int* in_sizes, int n_in,
                              void* d_out, int out_size, void* d_ws,
                              size_t ws_size, hipStream_t stream) {
  const float* x  = (const float*)d_in[0];
  const int*   ei = (const int*)d_in[1];
  const float* W1 = (const float*)d_in[2];
  const float* b1 = (const float*)d_in[3];
  const float* W2 = (const float*)d_in[4];
  const float* b2 = (const float*)d_in[5];
  const float* Wl = (const float*)d_in[6];
  const float* bl = (const float*)d_in[7];

  const int N = in_sizes[0] / D;
  const int E = in_sizes[1] / 2;

  float* logits = (float*)d_out;
  float* hcat   = logits + (size_t)N * NCLS;  // [N,256] output region
  float* agg    = (float*)d_ws;               // [N,64] scratch

  const int* src = ei;
  const int* dst = ei + E;

  const int nd = N * D;
  copy_x_kernel<<<(nd + 255) / 256, 256, 0, stream>>>(x, hcat, nd);

  const int tiles     = (N + 15) / 16;
  const int mlpBlocks = (tiles + MLP_WAVES - 1) / MLP_WAVES;
  const long long sthreads = (long long)E * 16;
  const int sBlocks = (int)((sthreads + 255) / 256);

  for (int l = 0; l < 3; ++l) {
    agg_init_kernel<<<(nd + 255) / 256, 256, 0, stream>>>(agg, hcat, D * l, nd);
    scatter_kernel<<<sBlocks, 256, 0, stream>>>(agg, hcat, D * l, src, dst, E);
    gin_mlp_kernel<<<mlpBlocks, MLP_WAVES * 32, 0, stream>>>(
        agg, W1 + (size_t)l * D * D, b1 + (size_t)l * D,
        W2 + (size_t)l * D * D, b2 + (size_t)l * D, hcat, D * (l + 1), N);
  }

  classifier_kernel<<<mlpBlocks, 128, 0, stream>>>(hcat, Wl, bl, logits, N);
}